// DeformCenterAttention_44667659878450
// MI455X (gfx1250) — compile-verified
//
#include <hip/hip_runtime.h>
#include <hip/hip_bf16.h>
#include <math.h>

// ---- problem constants (from reference) ----
#define BB   4
#define HH   64
#define WW   64
#define CC   128
#define GG   4
#define HC   32
#define K2   9
#define NPIX (HH*WW)          // 4096
#define MM   (BB*NPIX)        // 16384 rows
#define QKVC (3*CC)           // 384
#define SCALE 0.17677669529663689f  // 32^-0.5

typedef __attribute__((ext_vector_type(2))) float v2f;
typedef __attribute__((ext_vector_type(8))) float v8f;

// ---------------------------------------------------------------------------
// CDNA5 async copy: global -> LDS, per-lane, tracked by ASYNCcnt.
// lds_off is a raw LDS byte offset (VDST per ISA 10.x async addressing).
// ---------------------------------------------------------------------------
__device__ __forceinline__ void async_copy_b32(unsigned lds_off, const float* gptr) {
  asm volatile("global_load_async_to_lds_b32 %0, %1, off"
               :: "v"(lds_off), "v"(gptr) : "memory");
}
__device__ __forceinline__ void wait_async_all() {
  asm volatile("s_wait_asynccnt 0x0" ::: "memory");
}

// ---------------------------------------------------------------------------
// Shared WMMA GEMM: out[M x LDN] = A[M x 128] * W[128 x LDN] + bias.
// Block: 256 thr = 8 waves; block tile 128(M) x 32(N).
// B tile (128x32) staged once in LDS via async-to-LDS DMA, transposed with a
// padded stride (132) so the per-step ds_load_b64 fragment reads are
// bank-conflict-free. Each wave computes 16x32 (two 16x16 accumulators that
// share one A fragment) -> 2 WMMA per A load, 64 WMMA per wave.
// ---------------------------------------------------------------------------
template <int LDN>
__global__ __launch_bounds__(256) void gemm_wmma_k(const float* __restrict__ A,
                                                   const float* __restrict__ Wb,
                                                   const float* __restrict__ bias,
                                                   float* __restrict__ out) {
  __shared__ float Bs[32][132];                 // Bs[n][k], only LDS object -> base 0
  const int t    = threadIdx.x;
  const int lane = t & 31;
  const int wv   = t >> 5;
  const int bm   = blockIdx.x * 128;
  const int bn   = blockIdx.y * 32;

  // Stage B tile: element (k, n) -> Bs[n][k]; global reads coalesced in n.
  for (int i = 0; i < 16; ++i) {
    const int idx = t + 256 * i;                // 4096 elements
    const int k = idx >> 5, n = idx & 31;
    const unsigned lds_off = (unsigned)((n * 132 + k) * 4);
    async_copy_b32(lds_off, Wb + (size_t)k * LDN + bn + n);
  }
  wait_async_all();
  __syncthreads();

  const int tm  = bm + wv * 16;
  const int lm  = lane & 15;
  const int lhi = lane >> 4;                    // lanes>=16 supply K=k0+2,k0+3
  v8f acc0 = {}, acc1 = {};
  const float* arow = A + (size_t)(tm + lm) * CC;
  for (int k0 = 0; k0 < CC; k0 += 4) {
    const int kb = k0 + 2 * lhi;
    __builtin_prefetch(arow + kb + 16, 0, 3);   // global_prefetch_b8
    v2f a = *(const v2f*)(arow + kb);           // A(m, kb..kb+1)
    v2f b0, b1;
    b0.x = Bs[lm][kb];      b0.y = Bs[lm][kb + 1];       // cols bn..bn+15
    b1.x = Bs[lm + 16][kb]; b1.y = Bs[lm + 16][kb + 1];  // cols bn+16..bn+31
    acc0 = __builtin_amdgcn_wmma_f32_16x16x4_f32(false, a, false, b0,
                                                 (short)0, acc0, false, false);
    acc1 = __builtin_amdgcn_wmma_f32_16x16x4_f32(false, a, false, b1,
                                                 (short)0, acc1, false, false);
  }
  const float bias0 = bias[bn + lm];
  const float bias1 = bias[bn + 16 + lm];
  for (int r = 0; r < 8; ++r) {
    const int m = tm + r + 8 * lhi;             // C/D VGPR layout per ISA
    out[(size_t)m * LDN + bn + lm]      = acc0[r] + bias0;
    out[(size_t)m * LDN + bn + 16 + lm] = acc1[r] + bias1;
  }
}

// ---------------------------------------------------------------------------
// Depthwise 3x3 conv (pad 1) + bias, vectorized over 4 channels (float4).
// ---------------------------------------------------------------------------
__global__ void dwconv_k(const float* __restrict__ qkv,
                         const float* __restrict__ wdw,
                         const float* __restrict__ bdw,
                         float* __restrict__ out) {
  const int id = blockIdx.x * blockDim.x + threadIdx.x;   // over MM * 96
  if (id >= MM * (QKVC / 4)) return;
  const int c4  = id % (QKVC / 4);
  const int row = id / (QKVC / 4);
  const int pix = row & (NPIX - 1);
  const int b   = row >> 12;
  const int h   = pix >> 6, w = pix & 63;
  const int ch  = c4 * 4;
  float4 s = make_float4(bdw[ch], bdw[ch + 1], bdw[ch + 2], bdw[ch + 3]);
  for (int ky = 0; ky < 3; ++ky) {
    const int y = h + ky - 1;
    if ((unsigned)y >= (unsigned)HH) continue;
    for (int kx = 0; kx < 3; ++kx) {
      const int x = w + kx - 1;
      if ((unsigned)x >= (unsigned)WW) continue;
      const int tap = ky * 3 + kx;
      const float4 v = *(const float4*)(qkv +
          (size_t)((b << 12) + (y << 6) + x) * QKVC + ch);
      s.x += wdw[(ch + 0) * 9 + tap] * v.x;
      s.y += wdw[(ch + 1) * 9 + tap] * v.y;
      s.z += wdw[(ch + 2) * 9 + tap] * v.z;
      s.w += wdw[(ch + 3) * 9 + tap] * v.w;
    }
  }
  *(float4*)(out + (size_t)row * QKVC + ch) = s;
}

// ---------------------------------------------------------------------------
// Fold 1x1 weights through the 3x3 pconv weights:
//   u[slot, tap, ci] = sum_co w1x1[slot, co] * w3x3[co, ci, tap]
// slot 0..71 = (g,o) offset channels; slot 72..80 = mask channels.
// ---------------------------------------------------------------------------
__global__ void fold_w_k(const float* __restrict__ pow_, const float* __restrict__ woff,
                         const float* __restrict__ pmw,  const float* __restrict__ wmask,
                         float* __restrict__ u) {
  const int id = blockIdx.x * blockDim.x + threadIdx.x;   // slot*288 + tap*32 + ci
  if (id >= 81 * 288) return;
  const int slot = id / 288;
  const int rem  = id % 288;
  const int tap  = rem >> 5, ci = rem & 31;
  float s = 0.f;
  if (slot < 72) {
    const int g = slot / 18, o = slot % 18;
    for (int co = 0; co < 32; ++co)
      s += woff[(size_t)(g * 18 + o) * CC + co] *
           pow_[(size_t)((g * 32 + co) * 32 + ci) * 9 + tap];
  } else {
    const int o = slot - 72;
    for (int co = 0; co < 32; ++co)
      s += wmask[(size_t)o * CC + co] * pmw[(size_t)(co * 32 + ci) * 9 + tap];
  }
  u[id] = s;
}

// ---------------------------------------------------------------------------
// Per-pixel offsets (72 values) + sigmoid mask (9 values).
// One block per pixel; q center channels + 3x3 neighborhood (first 32 ch)
// staged in LDS; 81 threads each produce one output channel.
// ---------------------------------------------------------------------------
__global__ __launch_bounds__(128) void offmask_k(const float* __restrict__ qkv_dw,
                                                 const float* __restrict__ woff,
                                                 const float* __restrict__ boff,
                                                 const float* __restrict__ wmask,
                                                 const float* __restrict__ bmask,
                                                 const float* __restrict__ u,
                                                 float* __restrict__ om) {
  __shared__ float qc[CC];
  __shared__ float qn[9][32];
  const int rowp = blockIdx.x;            // b*4096 + pix
  const int pix  = rowp & (NPIX - 1);
  const int b    = rowp >> 12;
  const int h    = pix >> 6, w = pix & 63;
  const int t    = threadIdx.x;
  qc[t] = qkv_dw[(size_t)rowp * QKVC + t];
  for (int i = t; i < 288; i += 128) {
    const int tap = i >> 5, ci = i & 31;
    const int y = h + tap / 3 - 1, x = w + tap % 3 - 1;
    float v = 0.f;
    if ((unsigned)y < (unsigned)HH && (unsigned)x < (unsigned)WW)
      v = qkv_dw[(size_t)((b << 12) + (y << 6) + x) * QKVC + ci];
    qn[tap][ci] = v;
  }
  __syncthreads();
  if (t < 81) {
    float acc = (t < 72) ? boff[t] : bmask[t - 72];
    const float* wr = (t < 72) ? (woff + (size_t)t * CC) : (wmask + (size_t)(t - 72) * CC);
    for (int c = 32; c < CC; ++c) acc += wr[c] * qc[c];        // passthrough chans
    const float* up = u + t * 288;
    for (int tap = 0; tap < 9; ++tap)
      for (int ci = 0; ci < 32; ++ci)
        acc += up[tap * 32 + ci] * qn[tap][ci];                // folded 3x3 conv
    if (t >= 72) acc = 1.f / (1.f + __expf(-acc));             // sigmoid mask
    om[(size_t)rowp * 81 + t] = acc;
  }
}

// ---------------------------------------------------------------------------
// Deformable bilinear sample + 9-tap softmax attention.
// One wave32 per (b, g, pixel); lane = head channel (HC=32).
// Gathers are 128B coalesced thanks to channel-last layout.
// ---------------------------------------------------------------------------
__global__ __launch_bounds__(256) void deform_attn_k(const float* __restrict__ qkv_dw,
                                                     const float* __restrict__ om,
                                                     float* __restrict__ attn_out) {
  const int wid  = blockIdx.x * (blockDim.x >> 5) + (threadIdx.x >> 5);
  const int lane = threadIdx.x & 31;
  const int pix  = wid & (NPIX - 1);
  const int g    = (wid >> 12) & 3;
  const int b    = wid >> 14;
  const int h    = pix >> 6, w = pix & 63;
  const int row  = (b << 12) + pix;
  const int dil  = 2 * g + 1;
  const float q  = qkv_dw[(size_t)row * QKVC + g * HC + lane] * SCALE;
  const float* omr = om + (size_t)row * 81;
  float vs[9], lg[9];
  for (int tap = 0; tap < 9; ++tap) {
    const float dy = omr[g * 18 + tap * 2];
    const float dx = omr[g * 18 + tap * 2 + 1];
    const float m  = omr[72 + tap];
    const float py = (float)h + (float)((tap / 3) * dil - dil) + dy;
    const float px = (float)w + (float)((tap % 3) * dil - dil) + dx;
    const float y0f = floorf(py), x0f = floorf(px);
    const float wy = py - y0f, wx = px - x0f;
    const int y0 = (int)y0f, x0 = (int)x0f;
    float ks = 0.f, vv = 0.f;
    for (int cy = 0; cy < 2; ++cy) {
      const int yy = y0 + cy;
      if ((unsigned)yy >= (unsigned)HH) continue;
      const float wyc = cy ? wy : (1.f - wy);
      for (int cx = 0; cx < 2; ++cx) {
        const int xx = x0 + cx;
        if ((unsigned)xx >= (unsigned)WW) continue;
        const float wxc = cx ? wx : (1.f - wx);
        const float* p = qkv_dw +
            ((size_t)((b << 12) + (yy << 6) + xx) * QKVC + CC + g * HC + lane);
        const float wgt = wyc * wxc;
        ks += wgt * p[0];        // K channel
        vv += wgt * p[CC];       // V channel (+128)
      }
    }
    ks *= m; vv *= m;
    float s = q * ks;                                  // partial logit
    for (int o = 16; o > 0; o >>= 1) s += __shfl_xor(s, o, 32);
    lg[tap] = s;
    vs[tap] = vv;
  }
  float mx = lg[0];
  for (int tp = 1; tp < 9; ++tp) mx = fmaxf(mx, lg[tp]);
  float e[9], se = 0.f;
  for (int tp = 0; tp < 9; ++tp) { e[tp] = __expf(lg[tp] - mx); se += e[tp]; }
  const float inv = 1.f / se;
  float outc = 0.f;
  for (int tp = 0; tp < 9; ++tp) outc += e[tp] * inv * vs[tp];
  attn_out[(size_t)row * CC + g * HC + lane] = outc;
}

// ---------------------------------------------------------------------------
extern "C" void kernel_launch(void* const* d_in, const int* in_sizes, int n_in,
                              void* d_out, int out_size, void* d_ws, size_t ws_size,
                              hipStream_t stream) {
  const float* x            = (const float*)d_in[0];
  const float* w_qkv        = (const float*)d_in[1];
  const float* b_qkv        = (const float*)d_in[2];
  const float* w_dw         = (const float*)d_in[3];
  const float* b_dw         = (const float*)d_in[4];
  const float* pconv_off_w  = (const float*)d_in[5];
  const float* w_off        = (const float*)d_in[6];
  const float* b_off        = (const float*)d_in[7];
  const float* pconv_mask_w = (const float*)d_in[8];
  const float* w_mask       = (const float*)d_in[9];
  const float* b_mask       = (const float*)d_in[10];
  const float* w_proj       = (const float*)d_in[11];
  const float* b_proj       = (const float*)d_in[12];
  float* out = (float*)d_out;

  // workspace layout (floats). attn aliases qkv_lin (dead after dwconv).
  float* ws      = (float*)d_ws;
  float* qkv_lin = ws;                                   // M*384
  float* qkv_dw  = qkv_lin + (size_t)MM * QKVC;          // M*384
  float* om      = qkv_dw  + (size_t)MM * QKVC;          // M*81
  float* u       = om      + (size_t)MM * 81;            // 81*288
  float* attn    = qkv_lin;                              // reuse: M*128

  gemm_wmma_k<QKVC><<<dim3(MM / 128, QKVC / 32), 256, 0, stream>>>(x, w_qkv, b_qkv,
                                                                   qkv_lin);
  dwconv_k<<<(MM * (QKVC / 4) + 255) / 256, 256, 0, stream>>>(qkv_lin, w_dw, b_dw,
                                                              qkv_dw);
  fold_w_k<<<(81 * 288 + 255) / 256, 256, 0, stream>>>(pconv_off_w, w_off,
                                                       pconv_mask_w, w_mask, u);
  offmask_k<<<MM, 128, 0, stream>>>(qkv_dw, w_off, b_off, w_mask, b_mask, u, om);
  deform_attn_k<<<(BB * GG * NPIX) / 8, 256, 0, stream>>>(qkv_dw, om, attn);
  gemm_wmma_k<CC><<<dim3(MM / 128, CC / 32), 256, 0, stream>>>(attn, w_proj, b_proj,
                                                               out);
}